// GAT_103079215715
// MI455X (gfx1250) — compile-verified
//
#include <hip/hip_runtime.h>
#include <math.h>

#define HEADS 4
#define NEG_SLOPE 0.2f
#define GAT_EPS 1e-16f

typedef __attribute__((ext_vector_type(2))) float v2f;
typedef __attribute__((ext_vector_type(8))) float v8f;

// ---------------------------------------------------------------------------
// WMMA helpers: V_WMMA_F32_16X16X4_F32  (D = A(16x4) * B(4x16) + C(16x16))
// A layout: lanes 0-15 hold M=0..15 K={0,1}; lanes 16-31 hold M=0..15 K={2,3}
// B layout: lanes 0-15 hold N=0..15 K={0,1}; lanes 16-31 hold N=0..15 K={2,3}
// C/D layout: VGPR r, lane l -> (M = r + 8*(l>>4), N = l&15)
// ---------------------------------------------------------------------------
__device__ __forceinline__ v8f wmma_f32_16x16x4(v2f a, v2f b, v8f c) {
  return __builtin_amdgcn_wmma_f32_16x16x4_f32(false, a, false, b, (short)0, c,
                                               false, false);
}

__device__ __forceinline__ v2f load_frag_a(const float* __restrict__ A, int lda,
                                           int k0, int lane) {
  const int m = lane & 15;
  const int kk = (lane >> 4) << 1;
  v2f a;
  a.x = A[m * lda + k0 + kk];
  a.y = A[m * lda + k0 + kk + 1];
  return a;
}

__device__ __forceinline__ v2f load_frag_b(const float* __restrict__ B, int ldb,
                                           int k0, int col0, int lane) {
  const int n = lane & 15;
  const int kk = (lane >> 4) << 1;
  v2f b;
  b.x = B[(k0 + kk) * ldb + col0 + n];
  b.y = B[(k0 + kk + 1) * ldb + col0 + n];
  return b;
}

__device__ __forceinline__ float leaky(float v) {
  return v > 0.f ? v : NEG_SLOPE * v;
}

// float atomic max via signed/unsigned int ordering trick (init = -inf)
__device__ __forceinline__ void atomic_max_f32(float* addr, float value) {
  if (value >= 0.f)
    atomicMax((int*)addr, __float_as_int(value));
  else
    atomicMin((unsigned int*)addr, __float_as_uint(value));
}

// ---------------------------------------------------------------------------
// Encoder: h0 = relu(x @ W1 + b1) @ W2 + b2        x:[N,7] -> h0:[N,64]
// One block = 256 thr = 8 waves = 128 rows. Layer1 scalar (K=7), layer2 WMMA
// with all 4 column tiles accumulated per K-step (A fragment loaded once).
// ---------------------------------------------------------------------------
__global__ void __launch_bounds__(256) encoder_kernel(
    const float* __restrict__ x, const float* __restrict__ w1,
    const float* __restrict__ b1, const float* __restrict__ w2,
    const float* __restrict__ b2, float* __restrict__ h0, int n_nodes) {
  __shared__ float s_w1[7 * 64];
  __shared__ float s_b1[64];
  __shared__ float s_w2[64 * 64];
  __shared__ float s_b2[64];
  __shared__ float s_x[8][16][8];
  __shared__ float s_t[8][16][65];  // pad ld=65 to spread LDS banks

  const int tid = threadIdx.x;
  const int wave = tid >> 5, lane = tid & 31;
  for (int i = tid; i < 7 * 64; i += 256) s_w1[i] = w1[i];
  for (int i = tid; i < 64 * 64; i += 256) s_w2[i] = w2[i];
  if (tid < 64) { s_b1[tid] = b1[tid]; s_b2[tid] = b2[tid]; }

  const int row0 = blockIdx.x * 128 + wave * 16;
  const bool active = row0 < n_nodes;
  if (active && lane < 16) {
    #pragma unroll
    for (int k = 0; k < 7; ++k)
      s_x[wave][lane][k] = x[(size_t)(row0 + lane) * 7 + k];
  }
  __syncthreads();

  if (active) {
    for (int i = lane; i < 16 * 64; i += 32) {
      const int m = i >> 6, c = i & 63;
      float acc = s_b1[c];
      #pragma unroll
      for (int k = 0; k < 7; ++k) acc += s_x[wave][m][k] * s_w1[k * 64 + c];
      s_t[wave][m][c] = acc > 0.f ? acc : 0.f;
    }
  }
  __syncthreads();

  if (active) {
    const float* At = &s_t[wave][0][0];
    v8f acc[4];
    #pragma unroll
    for (int t = 0; t < 4; ++t)
      #pragma unroll
      for (int r = 0; r < 8; ++r) acc[t][r] = 0.f;  // inline-0 SRC2 on 1st wmma
    #pragma unroll
    for (int k0 = 0; k0 < 64; k0 += 4) {
      const v2f a = load_frag_a(At, 65, k0, lane);
      #pragma unroll
      for (int t = 0; t < 4; ++t)
        acc[t] = wmma_f32_16x16x4(a, load_frag_b(s_w2, 64, k0, t * 16, lane),
                                  acc[t]);
    }
    const int n = lane & 15, mbase = (lane >> 4) << 3;
    #pragma unroll
    for (int t = 0; t < 4; ++t) {
      const float bv = s_b2[t * 16 + n];
      #pragma unroll
      for (int r = 0; r < 8; ++r)
        h0[(size_t)(row0 + mbase + r) * 64 + t * 16 + n] = acc[t][r] + bv;
    }
  }
}

// ---------------------------------------------------------------------------
// Generic WMMA GEMM: C[M,NC] = A[M,K] @ B[K,NC]; B staged whole in LDS.
// 4 column tiles (64 cols) accumulated per K-step so each A fragment feeds
// 4 WMMAs.
// ---------------------------------------------------------------------------
template <int K, int NC>
__global__ void __launch_bounds__(256) gemm_kernel(const float* __restrict__ A,
                                                   const float* __restrict__ B,
                                                   float* __restrict__ C,
                                                   int m_rows) {
  __shared__ float s_B[K * NC];  // 64 KB for both instantiations
  const int tid = threadIdx.x;
  for (int i = tid; i < K * NC; i += 256) s_B[i] = B[i];
  __syncthreads();

  const int wave = tid >> 5, lane = tid & 31;
  const int row0 = blockIdx.x * 128 + wave * 16;
  if (row0 >= m_rows) return;
  const float* Ar = A + (size_t)row0 * K;
  const int n = lane & 15, mbase = (lane >> 4) << 3;

  for (int g = 0; g < NC / 64; ++g) {
    v8f acc[4];
    #pragma unroll
    for (int t = 0; t < 4; ++t)
      #pragma unroll
      for (int r = 0; r < 8; ++r) acc[t][r] = 0.f;
    #pragma unroll 4
    for (int k0 = 0; k0 < K; k0 += 4) {
      const v2f a = load_frag_a(Ar, K, k0, lane);
      #pragma unroll
      for (int t = 0; t < 4; ++t)
        acc[t] = wmma_f32_16x16x4(
            a, load_frag_b(s_B, NC, k0, g * 64 + t * 16, lane), acc[t]);
    }
    #pragma unroll
    for (int t = 0; t < 4; ++t)
      #pragma unroll
      for (int r = 0; r < 8; ++r)
        C[(size_t)(row0 + mbase + r) * NC + g * 64 + t * 16 + n] = acc[t][r];
  }
}

// ---------------------------------------------------------------------------
// alpha_src/alpha_dst per (node, head): dot(h[n,h,:], a[h,:])
// ---------------------------------------------------------------------------
template <int C>
__global__ void alpha_kernel(const float* __restrict__ h,
                             const float* __restrict__ a_src,
                             const float* __restrict__ a_dst,
                             float* __restrict__ aS, float* __restrict__ aD,
                             int n_nodes) {
  const int i = blockIdx.x * blockDim.x + threadIdx.x;
  if (i >= n_nodes * HEADS) return;
  const int n = i >> 2, hd = i & 3;
  const float* hp = h + (size_t)n * (HEADS * C) + hd * C;
  const float* asp = a_src + hd * C;
  const float* adp = a_dst + hd * C;
  float s = 0.f, d = 0.f;
  #pragma unroll 4
  for (int c = 0; c < C; ++c) {
    const float v = hp[c];
    s += v * asp[c];
    d += v * adp[c];
  }
  aS[i] = s;
  aD[i] = d;
}

__global__ void fill_kernel(float* __restrict__ p, float v, size_t n) {
  const size_t i = (size_t)blockIdx.x * blockDim.x + threadIdx.x;
  if (i < n) p[i] = v;
}

// ---------------------------------------------------------------------------
// Edge softmax pass 1: segment max over destination (self-loops appended)
// ---------------------------------------------------------------------------
__global__ void edge_max_kernel(const int* __restrict__ src,
                                const int* __restrict__ dst,
                                const float* __restrict__ aS,
                                const float* __restrict__ aD,
                                float* __restrict__ mx, int n_edges,
                                int n_nodes) {
  const int e = blockIdx.x * blockDim.x + threadIdx.x;
  if (e >= n_edges + n_nodes) return;
  const int s = e < n_edges ? src[e] : e - n_edges;
  const int d = e < n_edges ? dst[e] : e - n_edges;
  const float4 s4 = *(const float4*)(aS + (size_t)s * 4);
  const float4 d4 = *(const float4*)(aD + (size_t)d * 4);
  float* mp = mx + (size_t)d * 4;
  atomic_max_f32(mp + 0, leaky(s4.x + d4.x));
  atomic_max_f32(mp + 1, leaky(s4.y + d4.y));
  atomic_max_f32(mp + 2, leaky(s4.z + d4.z));
  atomic_max_f32(mp + 3, leaky(s4.w + d4.w));
}

// Edge softmax pass 2: denom = segment_sum(exp(e - m))
__global__ void edge_sum_kernel(const int* __restrict__ src,
                                const int* __restrict__ dst,
                                const float* __restrict__ aS,
                                const float* __restrict__ aD,
                                const float* __restrict__ mx,
                                float* __restrict__ den, int n_edges,
                                int n_nodes) {
  const int e = blockIdx.x * blockDim.x + threadIdx.x;
  if (e >= n_edges + n_nodes) return;
  const int s = e < n_edges ? src[e] : e - n_edges;
  const int d = e < n_edges ? dst[e] : e - n_edges;
  const float4 s4 = *(const float4*)(aS + (size_t)s * 4);
  const float4 d4 = *(const float4*)(aD + (size_t)d * 4);
  const float4 m4 = *(const float4*)(mx + (size_t)d * 4);
  float* dp = den + (size_t)d * 4;
  atomicAdd(dp + 0, __expf(leaky(s4.x + d4.x) - m4.x));
  atomicAdd(dp + 1, __expf(leaky(s4.y + d4.y) - m4.y));
  atomicAdd(dp + 2, __expf(leaky(s4.z + d4.z) - m4.z));
  atomicAdd(dp + 3, __expf(leaky(s4.w + d4.w) - m4.w));
}

// ---------------------------------------------------------------------------
// Pass 3: out[dst] += h[src] * softmax_weight. One wave per edge, each lane
// owns CH/32 contiguous channels (within one head). h gathers hit L2 (192MB).
// Explicit b128/b64 gathers; scatter via global_atomic_add_f32.
// ---------------------------------------------------------------------------
template <int C>
__global__ void __launch_bounds__(256) edge_aggregate_kernel(
    const int* __restrict__ src, const int* __restrict__ dst,
    const float* __restrict__ h, const float* __restrict__ aS,
    const float* __restrict__ aD, const float* __restrict__ mx,
    const float* __restrict__ den, float* __restrict__ out, int n_edges,
    int n_nodes) {
  constexpr int CH = HEADS * C;
  constexpr int PER = CH / 32;
  const int gt = blockIdx.x * blockDim.x + threadIdx.x;
  const int e = gt >> 5;
  const int lane = threadIdx.x & 31;
  if (e >= n_edges + n_nodes) return;
  const int s = e < n_edges ? src[e] : e - n_edges;
  const int d = e < n_edges ? dst[e] : e - n_edges;
  const int c0 = lane * PER;
  const int hd = c0 / C;
  const float ev = leaky(aS[(size_t)s * 4 + hd] + aD[(size_t)d * 4 + hd]);
  const float w =
      __expf(ev - mx[(size_t)d * 4 + hd]) / (den[(size_t)d * 4 + hd] + GAT_EPS);
  const float* hp = h + (size_t)s * CH + c0;
  float* op = out + (size_t)d * CH + c0;
  float v[PER];
  if constexpr (PER == 8) {
    const float4 a = *(const float4*)hp;
    const float4 b = *(const float4*)(hp + 4);
    v[0] = a.x; v[1] = a.y; v[2] = a.z; v[3] = a.w;
    v[4] = b.x; v[5] = b.y; v[6] = b.z; v[7] = b.w;
  } else {
    const float2 a = *(const float2*)hp;
    v[0] = a.x; v[1] = a.y;
  }
  #pragma unroll
  for (int i = 0; i < PER; ++i) atomicAdd(op + i, v[i] * w);
}

// h = elu(h + bias[c]) in place  (ch must be power of two)
__global__ void bias_elu_kernel(float* __restrict__ h,
                                const float* __restrict__ bias, size_t n,
                                int cmask) {
  const size_t i = (size_t)blockIdx.x * blockDim.x + threadIdx.x;
  if (i >= n) return;
  const float v = h[i] + bias[i & cmask];
  h[i] = v > 0.f ? v : __expf(v) - 1.f;
}

// ---------------------------------------------------------------------------
// Decoder: out = relu((h + g2_b) @ W1 + b1) @ W2 + b2   [N,64] -> [N,4]
// Layer1 via WMMA (GAT2 bias folded into A-fragment load, loaded once per
// K-step for all 4 column tiles); 64->4 tail scalar.
// ---------------------------------------------------------------------------
__global__ void __launch_bounds__(256) decoder_kernel(
    const float* __restrict__ h, const float* __restrict__ gbias,
    const float* __restrict__ w1, const float* __restrict__ b1,
    const float* __restrict__ w2, const float* __restrict__ b2,
    float* __restrict__ out, int n_nodes) {
  __shared__ float s_w1[64 * 64];
  __shared__ float s_b1[64];
  __shared__ float s_gb[64];
  __shared__ float s_w2[64 * 4];
  __shared__ float s_b2[4];
  __shared__ float s_t[8][16][65];

  const int tid = threadIdx.x;
  for (int i = tid; i < 64 * 64; i += 256) s_w1[i] = w1[i];
  for (int i = tid; i < 64 * 4; i += 256) s_w2[i] = w2[i];
  if (tid < 64) { s_b1[tid] = b1[tid]; s_gb[tid] = gbias[tid]; }
  if (tid < 4) s_b2[tid] = b2[tid];
  __syncthreads();

  const int wave = tid >> 5, lane = tid & 31;
  const int row0 = blockIdx.x * 128 + wave * 16;
  const bool active = row0 < n_nodes;
  if (active) {
    const float* Ar = h + (size_t)row0 * 64;
    const int m = lane & 15;
    const int kk = (lane >> 4) << 1;
    v8f acc[4];
    #pragma unroll
    for (int t = 0; t < 4; ++t)
      #pragma unroll
      for (int r = 0; r < 8; ++r) acc[t][r] = 0.f;
    #pragma unroll
    for (int k0 = 0; k0 < 64; k0 += 4) {
      v2f a;
      a.x = Ar[m * 64 + k0 + kk] + s_gb[k0 + kk];
      a.y = Ar[m * 64 + k0 + kk + 1] + s_gb[k0 + kk + 1];
      #pragma unroll
      for (int t = 0; t < 4; ++t)
        acc[t] = wmma_f32_16x16x4(a, load_frag_b(s_w1, 64, k0, t * 16, lane),
                                  acc[t]);
    }
    const int n = lane & 15, mbase = (lane >> 4) << 3;
    #pragma unroll
    for (int t = 0; t < 4; ++t) {
      const float bv = s_b1[t * 16 + n];
      #pragma unroll
      for (int r = 0; r < 8; ++r) {
        const float v = acc[t][r] + bv;
        s_t[wave][mbase + r][t * 16 + n] = v > 0.f ? v : 0.f;
      }
    }
  }
  __syncthreads();
  if (active) {
    #pragma unroll
    for (int t2 = 0; t2 < 2; ++t2) {
      const int task = lane * 2 + t2;  // 16 rows x 4 cols = 64 tasks
      const int m = task >> 2, j = task & 3;
      float acc = s_b2[j];
      #pragma unroll 8
      for (int k = 0; k < 64; ++k) acc += s_t[wave][m][k] * s_w2[k * 4 + j];
      out[(size_t)(row0 + m) * 4 + j] = acc;
    }
  }
}

// ---------------------------------------------------------------------------
extern "C" void kernel_launch(void* const* d_in, const int* in_sizes, int n_in,
                              void* d_out, int out_size, void* d_ws,
                              size_t ws_size, hipStream_t stream) {
  (void)n_in; (void)out_size; (void)ws_size;
  const float* x      = (const float*)d_in[0];
  const int*   ei     = (const int*)d_in[1];
  const float* enc_w1 = (const float*)d_in[2];
  const float* enc_b1 = (const float*)d_in[3];
  const float* enc_w2 = (const float*)d_in[4];
  const float* enc_b2 = (const float*)d_in[5];
  const float* g1_w   = (const float*)d_in[6];
  const float* g1_as  = (const float*)d_in[7];
  const float* g1_ad  = (const float*)d_in[8];
  const float* g1_b   = (const float*)d_in[9];
  const float* g2_w   = (const float*)d_in[10];
  const float* g2_as  = (const float*)d_in[11];
  const float* g2_ad  = (const float*)d_in[12];
  const float* g2_b   = (const float*)d_in[13];
  const float* dec_w1 = (const float*)d_in[14];
  const float* dec_b1 = (const float*)d_in[15];
  const float* dec_w2 = (const float*)d_in[16];
  const float* dec_b2 = (const float*)d_in[17];

  const int N = in_sizes[0] / 7;
  const int E = in_sizes[1] / 2;
  const int* src = ei;
  const int* dst = ei + E;
  const int ET = E + N;  // edges + self loops

  char* ws = (char*)d_ws;
  size_t off = 0;
  auto alloc = [&](size_t bytes) -> float* {
    float* p = (float*)(ws + off);
    off = (off + bytes + 255) & ~(size_t)255;
    return p;
  };
  float* h0   = alloc((size_t)N * 64 * 4);   // encoder output
  float* h1   = alloc((size_t)N * 256 * 4);  // GAT1 projection
  float* out1 = alloc((size_t)N * 256 * 4);  // GAT1 aggregate -> elu in place
  float* h2p  = alloc((size_t)N * 64 * 4);   // GAT2 projection
  float* out2 = alloc((size_t)N * 64 * 4);   // GAT2 aggregate (pre-bias)
  float* aS1  = alloc((size_t)N * 4 * 4);
  float* aD1  = alloc((size_t)N * 4 * 4);
  float* mx1  = alloc((size_t)N * 4 * 4);
  float* dn1  = alloc((size_t)N * 4 * 4);
  float* aS2  = alloc((size_t)N * 4 * 4);
  float* aD2  = alloc((size_t)N * 4 * 4);
  float* mx2  = alloc((size_t)N * 4 * 4);
  float* dn2  = alloc((size_t)N * 4 * 4);

  const dim3 blk(256);
  const int rowBlocks = (N + 127) / 128;
  const int edgeBlocks = (ET + 255) / 256;
  const int aggBlocks = (ET * 32 + 255) / 256;
  auto fb = [](size_t n) { return (unsigned)((n + 255) / 256); };

  // encoder
  encoder_kernel<<<rowBlocks, blk, 0, stream>>>(x, enc_w1, enc_b1, enc_w2,
                                                enc_b2, h0, N);
  // ---- GAT layer 1 ----
  gemm_kernel<64, 256><<<rowBlocks, blk, 0, stream>>>(h0, g1_w, h1, N);
  alpha_kernel<64><<<(N * 4 + 255) / 256, blk, 0, stream>>>(h1, g1_as, g1_ad,
                                                            aS1, aD1, N);
  fill_kernel<<<fb((size_t)N * 4), blk, 0, stream>>>(mx1, -INFINITY, (size_t)N * 4);
  fill_kernel<<<fb((size_t)N * 4), blk, 0, stream>>>(dn1, 0.f, (size_t)N * 4);
  fill_kernel<<<fb((size_t)N * 256), blk, 0, stream>>>(out1, 0.f, (size_t)N * 256);
  edge_max_kernel<<<edgeBlocks, blk, 0, stream>>>(src, dst, aS1, aD1, mx1, E, N);
  edge_sum_kernel<<<edgeBlocks, blk, 0, stream>>>(src, dst, aS1, aD1, mx1, dn1, E, N);
  edge_aggregate_kernel<64><<<aggBlocks, blk, 0, stream>>>(
      src, dst, h1, aS1, aD1, mx1, dn1, out1, E, N);
  bias_elu_kernel<<<fb((size_t)N * 256), blk, 0, stream>>>(out1, g1_b,
                                                           (size_t)N * 256, 255);
  // ---- GAT layer 2 ----
  gemm_kernel<256, 64><<<rowBlocks, blk, 0, stream>>>(out1, g2_w, h2p, N);
  alpha_kernel<16><<<(N * 4 + 255) / 256, blk, 0, stream>>>(h2p, g2_as, g2_ad,
                                                            aS2, aD2, N);
  fill_kernel<<<fb((size_t)N * 4), blk, 0, stream>>>(mx2, -INFINITY, (size_t)N * 4);
  fill_kernel<<<fb((size_t)N * 4), blk, 0, stream>>>(dn2, 0.f, (size_t)N * 4);
  fill_kernel<<<fb((size_t)N * 64), blk, 0, stream>>>(out2, 0.f, (size_t)N * 64);
  edge_max_kernel<<<edgeBlocks, blk, 0, stream>>>(src, dst, aS2, aD2, mx2, E, N);
  edge_sum_kernel<<<edgeBlocks, blk, 0, stream>>>(src, dst, aS2, aD2, mx2, dn2, E, N);
  edge_aggregate_kernel<16><<<aggBlocks, blk, 0, stream>>>(
      src, dst, h2p, aS2, aD2, mx2, dn2, out2, E, N);
  // decoder (g2_b folded into A-fragment load)
  decoder_kernel<<<rowBlocks, blk, 0, stream>>>(out2, g2_b, dec_w1, dec_b1,
                                                dec_w2, dec_b2, (float*)d_out, N);
}